// ArithmeticWorldModel_30545807409197
// MI455X (gfx1250) — compile-verified
//
#include <hip/hip_runtime.h>
#include <hip/hip_bf16.h>
#include <math.h>

// ---------------------------------------------------------------------------
// ArithmeticWorldModel on MI455X (gfx1250, wave32).
// One wave / block handles a 16-row tile. All GEMMs use V_WMMA_F32_16X16X4_F32
// (fp32 in/out -> matches reference numerics). Hankel "soft rank" via WMMA
// Gram matrix + cyclic Jacobi eigen-sweeps in LDS. Scan exploits that only the
// z-column of `inp` changes per step (bases precomputed once).
// ---------------------------------------------------------------------------

typedef __attribute__((ext_vector_type(2))) float v2f;
typedef __attribute__((ext_vector_type(8))) float v8f;

struct Params {
  const float *ctx;
  const float *enc_w1, *enc_b1, *ln_g, *ln_b, *enc_w2, *enc_b2, *enc_w3, *enc_b3;
  const float *pe_thr;
  const float *add_w, *add_b, *mul_w, *mul_b;
  const float *sm_w1, *sm_b1, *sm_w2, *sm_b2, *sm_w3, *sm_b3;
  const float *pw_w1, *pw_b1, *pw_w2, *pw_b2, *pw_w3, *pw_b3;
  const float *gate_w1, *gate_b1, *gate_w2, *gate_b2;
  float *out;
};

__device__ __forceinline__ v8f wmma4(v2f a, v2f b, v8f c) {
  // D = A(16x4,f32) * B(4x16,f32) + C(16x16,f32)
  return __builtin_amdgcn_wmma_f32_16x16x4_f32(
      /*neg_a=*/false, a, /*neg_b=*/false, b,
      /*c_mod=*/(short)0, c, /*reuse_a=*/false, /*reuse_b=*/false);
}

// C(16 x [n0..n0+16)) = A_lds(16 x K, row-major lda) * W(K x ldw, row-major)
__device__ __forceinline__ v8f gemm_tile(const float* As, int lda,
                                         const float* W, int ldw,
                                         int K, int n0, int lane) {
  const int m = lane & 15;
  const int half = lane >> 4;
  v8f acc = {};
  for (int k0 = 0; k0 < K; k0 += 4) {
    const int k1 = k0 + (half ? 2 : 0);
    const int k2 = k0 + (half ? 3 : 1);
    v2f a, b;
    a.x = As[m * lda + k1];
    a.y = As[m * lda + k2];
    b.x = W[k1 * ldw + n0 + m];
    b.y = W[k2 * ldw + n0 + m];
    acc = wmma4(a, b, acc);
  }
  return acc;
}

// ---- LDS arena layout (floats). Total 16256 floats = 65,024 B. -------------
// Persistent (alive through the scan):
#define L_GATED   0       // 16x16
#define L_GVEC    256     // 16x4 softmax gate
#define L_SMBASE  320     // 16x64
#define L_PWBASE  1344    // 16x64
#define L_SMROW0  2368    // 64  (sm_w1 row 0)
#define L_PWROW0  2432    // 64
#define L_SMW2    2496    // 64x32
#define L_PWW2    4544    // 64x32
#define L_SMW3    6592    // 32
#define L_PWW3    6624    // 32
#define L_SMB2    6656    // 32
#define L_PWB2    6688    // 32
#define L_ZBUF    6720    // 16
#define L_CONF    6736    // 16
#define L_ABASE   6752    // 16
#define L_MBASE   6768    // 16
// Scratch:
#define L_CTX     6784    // 16x64 context tile
#define L_CODE    7808    // 16x16 code
#define L_R       8064    // 8192-float reusable region:
#define L_H1      (L_R)           // 16x128 (encoder)
#define L_H2      (L_R + 2048)    // 16x64  (encoder)
#define L_GS      (L_R)           // 8 x 32x32 Gram matrices
#define L_T1      (L_R)           // 16x64  (scan temp)
#define L_S2      (L_R + 1024)    // 16x32  (scan temp)
#define L_RELU1   (L_R + 3072)    // 16x32  (gate temp)
#define L_TOTAL   (L_R + 8192)

__global__ __launch_bounds__(32)
void awm_kernel(Params p) {
  __shared__ __align__(16) float sm[L_TOTAL];
  const int lane = threadIdx.x;
  const int m = lane & 15;
  const int half = lane >> 4;
  const int tile = blockIdx.x;

  if (lane == 0) {
    __builtin_prefetch(p.enc_w1, 0, 0);
    __builtin_prefetch(p.enc_w2, 0, 0);
    __builtin_prefetch(p.sm_w2, 0, 0);
    __builtin_prefetch(p.pw_w2, 0, 0);
  }

  // ---- load 16x64 context tile -------------------------------------------
  {
    const float4* src = (const float4*)(p.ctx + (size_t)tile * 16 * 64);
    float4* dst = (float4*)&sm[L_CTX];
    for (int i = lane; i < 256; i += 32) dst[i] = src[i];
  }
  __syncthreads();

  // ---- encoder layer 1: h1 = ctx @ enc_w1 + b1 ---------------------------
  for (int n0 = 0; n0 < 128; n0 += 16) {
    v8f acc = gemm_tile(&sm[L_CTX], 64, p.enc_w1, 128, 64, n0, lane);
    const float bv = p.enc_b1[n0 + m];
    for (int r = 0; r < 8; ++r)
      sm[L_H1 + (r + (half ? 8 : 0)) * 128 + n0 + m] = acc[r] + bv;
  }
  __syncthreads();

  // ---- layernorm + relu (lane r handles row r) ---------------------------
  if (lane < 16) {
    float mu = 0.f;
    for (int c = 0; c < 128; ++c) mu += sm[L_H1 + lane * 128 + c];
    mu *= (1.f / 128.f);
    float var = 0.f;
    for (int c = 0; c < 128; ++c) {
      float d = sm[L_H1 + lane * 128 + c] - mu;
      var += d * d;
    }
    var *= (1.f / 128.f);
    const float inv = rsqrtf(var + 1e-5f);
    for (int c = 0; c < 128; ++c) {
      float v = (sm[L_H1 + lane * 128 + c] - mu) * inv * p.ln_g[c] + p.ln_b[c];
      sm[L_H1 + lane * 128 + c] = v > 0.f ? v : 0.f;
    }
  }
  __syncthreads();

  // ---- encoder layer 2: h2 = relu(h1 @ enc_w2 + b2) ----------------------
  for (int n0 = 0; n0 < 64; n0 += 16) {
    v8f acc = gemm_tile(&sm[L_H1], 128, p.enc_w2, 64, 128, n0, lane);
    const float bv = p.enc_b2[n0 + m];
    for (int r = 0; r < 8; ++r) {
      float v = acc[r] + bv;
      sm[L_H2 + (r + (half ? 8 : 0)) * 64 + n0 + m] = v > 0.f ? v : 0.f;
    }
  }
  __syncthreads();

  // ---- code = h2 @ enc_w3 + b3 (16x16) -----------------------------------
  {
    v8f acc = gemm_tile(&sm[L_H2], 64, p.enc_w3, 16, 64, 0, lane);
    const float bv = p.enc_b3[m];
    for (int r = 0; r < 8; ++r)
      sm[L_CODE + (r + (half ? 8 : 0)) * 16 + m] = acc[r] + bv;
  }
  __syncthreads();

  // ---- soft Hankel rank: Gram (WMMA) + Jacobi eigen sweeps ---------------
  // H[d][c] = ctx[d+c], d<32, c<33; G = H H^T (32x32 SPD), sv = sqrt(eig).
  const float pe_thr = p.pe_thr[0];
  for (int batch = 0; batch < 2; ++batch) {
    for (int rr = 0; rr < 8; ++rr) {
      const float* crow = &sm[L_CTX + (batch * 8 + rr) * 64];
      for (int ti = 0; ti < 32; ti += 16)
        for (int tj = 0; tj < 32; tj += 16) {
          v8f acc = {};
          for (int k0 = 0; k0 < 36; k0 += 4) {  // K=33 padded to 36
            const int k1 = k0 + (half ? 2 : 0);
            const int k2 = k0 + (half ? 3 : 1);
            v2f a, b;
            a.x = (k1 < 33) ? crow[ti + m + k1] : 0.f;
            a.y = (k2 < 33) ? crow[ti + m + k2] : 0.f;
            b.x = (k1 < 33) ? crow[tj + m + k1] : 0.f;
            b.y = (k2 < 33) ? crow[tj + m + k2] : 0.f;
            acc = wmma4(a, b, acc);
          }
          for (int r = 0; r < 8; ++r) {
            const int i = ti + r + (half ? 8 : 0);
            sm[L_GS + rr * 1024 + i * 32 + tj + m] = acc[r];
          }
        }
    }
    __syncthreads();
    if (lane < 8) {
      float* G = &sm[L_GS + lane * 1024];
      for (int sweep = 0; sweep < 5; ++sweep) {
        for (int pp = 0; pp < 31; ++pp) {
          for (int q = pp + 1; q < 32; ++q) {
            const float apq = G[pp * 32 + q];
            if (fabsf(apq) < 1e-12f) continue;
            const float app = G[pp * 32 + pp];
            const float aqq = G[q * 32 + q];
            const float tau = (aqq - app) / (2.0f * apq);
            const float t =
                (tau >= 0.f ? 1.f : -1.f) / (fabsf(tau) + sqrtf(1.f + tau * tau));
            const float c = 1.0f / sqrtf(1.f + t * t);
            const float s = t * c;
#pragma clang loop unroll(disable)
            for (int j = 0; j < 32; ++j) {
              const float rp = G[pp * 32 + j], rq = G[q * 32 + j];
              G[pp * 32 + j] = c * rp - s * rq;
              G[q * 32 + j] = s * rp + c * rq;
            }
#pragma clang loop unroll(disable)
            for (int i = 0; i < 32; ++i) {
              const float cp = G[i * 32 + pp], cq = G[i * 32 + q];
              G[i * 32 + pp] = c * cp - s * cq;
              G[i * 32 + q] = s * cp + c * cq;
            }
          }
        }
      }
      float rank = 0.f;
      for (int i = 0; i < 32; ++i) {
        float lam = G[i * 32 + i];
        lam = lam > 0.f ? lam : 0.f;
        const float sv = sqrtf(lam);
        rank += sv / (sv + 1e-4f);
      }
      sm[L_CONF + batch * 8 + lane] = 1.f / (1.f + expf(-(rank - pe_thr)));
    }
    __syncthreads();
  }

  // ---- gated_code = confidence * code ------------------------------------
  for (int i = lane; i < 256; i += 32)
    sm[L_GATED + i] = sm[L_CONF + (i >> 4)] * sm[L_CODE + i];
  __syncthreads();

  // ---- gate g = softmax(relu(gated@gw1+b1)@gw2+b2) -----------------------
  for (int n0 = 0; n0 < 32; n0 += 16) {
    v8f acc = gemm_tile(&sm[L_GATED], 16, p.gate_w1, 32, 16, n0, lane);
    const float bv = p.gate_b1[n0 + m];
    for (int r = 0; r < 8; ++r) {
      float v = acc[r] + bv;
      sm[L_RELU1 + (r + (half ? 8 : 0)) * 32 + n0 + m] = v > 0.f ? v : 0.f;
    }
  }
  __syncthreads();
  if (lane < 16) {
    float a4[4];
    for (int n = 0; n < 4; ++n) {
      float a = p.gate_b2[n];
      for (int k = 0; k < 32; ++k)
        a += sm[L_RELU1 + lane * 32 + k] * p.gate_w2[k * 4 + n];
      a4[n] = a;
    }
    const float mx = fmaxf(fmaxf(a4[0], a4[1]), fmaxf(a4[2], a4[3]));
    const float e0 = expf(a4[0] - mx), e1 = expf(a4[1] - mx);
    const float e2 = expf(a4[2] - mx), e3 = expf(a4[3] - mx);
    const float si = 1.f / (e0 + e1 + e2 + e3);
    sm[L_GVEC + lane * 4 + 0] = e0 * si;
    sm[L_GVEC + lane * 4 + 1] = e1 * si;
    sm[L_GVEC + lane * 4 + 2] = e2 * si;
    sm[L_GVEC + lane * 4 + 3] = e3 * si;
  }

  // ---- step-invariant bases: gated @ W[1:,:] + b --------------------------
  for (int n0 = 0; n0 < 64; n0 += 16) {
    v8f acc = gemm_tile(&sm[L_GATED], 16, p.sm_w1 + 64, 64, 16, n0, lane);
    const float bv = p.sm_b1[n0 + m];
    for (int r = 0; r < 8; ++r)
      sm[L_SMBASE + (r + (half ? 8 : 0)) * 64 + n0 + m] = acc[r] + bv;
  }
  for (int n0 = 0; n0 < 64; n0 += 16) {
    v8f acc = gemm_tile(&sm[L_GATED], 16, p.pw_w1 + 64, 64, 16, n0, lane);
    const float bv = p.pw_b1[n0 + m];
    for (int r = 0; r < 8; ++r)
      sm[L_PWBASE + (r + (half ? 8 : 0)) * 64 + n0 + m] = acc[r] + bv;
  }

  // ---- stage scan weights into LDS ---------------------------------------
  for (int i = lane; i < 2048; i += 32) {
    sm[L_SMW2 + i] = p.sm_w2[i];
    sm[L_PWW2 + i] = p.pw_w2[i];
  }
  for (int i = lane; i < 64; i += 32) {
    sm[L_SMROW0 + i] = p.sm_w1[i];  // row 0 of (17x64)
    sm[L_PWROW0 + i] = p.pw_w1[i];
  }
  if (lane < 32) {
    sm[L_SMW3 + lane] = p.sm_w3[lane];
    sm[L_PWW3 + lane] = p.pw_w3[lane];
    sm[L_SMB2 + lane] = p.sm_b2[lane];
    sm[L_PWB2 + lane] = p.pw_b2[lane];
  }
  if (lane < 16) {
    float a = p.add_b[0], mb = p.mul_b[0];
    for (int c = 0; c < 16; ++c) {
      const float gc = sm[L_GATED + lane * 16 + c];
      a += gc * p.add_w[1 + c];
      mb += gc * p.mul_w[1 + c];
    }
    sm[L_ABASE + lane] = a;
    sm[L_MBASE + lane] = mb;
  }
  __syncthreads();

  // ---- scan: 64 steps -----------------------------------------------------
  float addw0 = 0.f, mulw0 = 0.f, smb3 = 0.f, pwb3 = 0.f, z = 0.f;
  float g0 = 0.f, g1 = 0.f, g2 = 0.f, g3 = 0.f;
  if (lane < 16) {
    addw0 = p.add_w[0];
    mulw0 = p.mul_w[0];
    smb3 = p.sm_b3[0];
    pwb3 = p.pw_b3[0];
    z = sm[L_CTX + lane * 64 + 63];  // z0 = context[:, -1]
    g0 = sm[L_GVEC + lane * 4 + 0];
    g1 = sm[L_GVEC + lane * 4 + 1];
    g2 = sm[L_GVEC + lane * 4 + 2];
    g3 = sm[L_GVEC + lane * 4 + 3];
  }
  float* outp = p.out + (size_t)tile * 16 * 64;

  for (int step = 0; step < 64; ++step) {
    if (lane < 16) sm[L_ZBUF + lane] = z;
    __syncthreads();

    // sm path: t1 = tanh(z*w1_row0 + base)
    for (int i = lane; i < 1024; i += 32) {
      const int r = i >> 6, k = i & 63;
      sm[L_T1 + i] = tanhf(sm[L_ZBUF + r] * sm[L_SMROW0 + k] + sm[L_SMBASE + i]);
    }
    __syncthreads();
    for (int n0 = 0; n0 < 32; n0 += 16) {
      v8f acc = gemm_tile(&sm[L_T1], 64, &sm[L_SMW2], 32, 64, n0, lane);
      const float bv = sm[L_SMB2 + n0 + m];
      for (int r = 0; r < 8; ++r)
        sm[L_S2 + (r + (half ? 8 : 0)) * 32 + n0 + m] = tanhf(acc[r] + bv);
    }
    __syncthreads();
    float h2v = 0.f;
    if (lane < 16) {
      h2v = smb3;
      for (int k = 0; k < 32; ++k)
        h2v += sm[L_S2 + lane * 32 + k] * sm[L_SMW3 + k];
    }
    __syncthreads();

    // pw path: t1 = relu(z*w1_row0 + base)
    for (int i = lane; i < 1024; i += 32) {
      const int r = i >> 6, k = i & 63;
      const float v = sm[L_ZBUF + r] * sm[L_PWROW0 + k] + sm[L_PWBASE + i];
      sm[L_T1 + i] = v > 0.f ? v : 0.f;
    }
    __syncthreads();
    for (int n0 = 0; n0 < 32; n0 += 16) {
      v8f acc = gemm_tile(&sm[L_T1], 64, &sm[L_PWW2], 32, 64, n0, lane);
      const float bv = sm[L_PWB2 + n0 + m];
      for (int r = 0; r < 8; ++r) {
        const float v = acc[r] + bv;
        sm[L_S2 + (r + (half ? 8 : 0)) * 32 + n0 + m] = v > 0.f ? v : 0.f;
      }
    }
    __syncthreads();

    if (lane < 16) {
      float h3v = pwb3;
      for (int k = 0; k < 32; ++k)
        h3v += sm[L_S2 + lane * 32 + k] * sm[L_PWW3 + k];
      const float h0v = addw0 * z + sm[L_ABASE + lane];
      const float zs = z + 1e-12f;
      const float sgn = zs > 0.f ? 1.f : (zs < 0.f ? -1.f : 0.f);
      const float lz = logf(fabsf(z) + 1e-8f) * sgn;
      float e = mulw0 * lz + sm[L_MBASE + lane];
      e = fminf(fmaxf(e, -20.f), 20.f);
      const float h1v = expf(e);
      z = g0 * h0v + g1 * h1v + g2 * h2v + g3 * h3v;
      outp[lane * 64 + step] = z;
    }
    __syncthreads();
  }
}

extern "C" void kernel_launch(void* const* d_in, const int* in_sizes, int n_in,
                              void* d_out, int out_size, void* d_ws, size_t ws_size,
                              hipStream_t stream) {
  (void)in_sizes; (void)n_in; (void)out_size; (void)d_ws; (void)ws_size;
  Params P;
  P.ctx = (const float*)d_in[0];
  // d_in[1] = n_steps (== 64, fixed by problem size)
  P.enc_w1 = (const float*)d_in[2];  P.enc_b1 = (const float*)d_in[3];
  P.ln_g   = (const float*)d_in[4];  P.ln_b   = (const float*)d_in[5];
  P.enc_w2 = (const float*)d_in[6];  P.enc_b2 = (const float*)d_in[7];
  P.enc_w3 = (const float*)d_in[8];  P.enc_b3 = (const float*)d_in[9];
  P.pe_thr = (const float*)d_in[10];
  P.add_w  = (const float*)d_in[11]; P.add_b  = (const float*)d_in[12];
  P.mul_w  = (const float*)d_in[13]; P.mul_b  = (const float*)d_in[14];
  P.sm_w1  = (const float*)d_in[15]; P.sm_b1  = (const float*)d_in[16];
  P.sm_w2  = (const float*)d_in[17]; P.sm_b2  = (const float*)d_in[18];
  P.sm_w3  = (const float*)d_in[19]; P.sm_b3  = (const float*)d_in[20];
  P.pw_w1  = (const float*)d_in[21]; P.pw_b1  = (const float*)d_in[22];
  P.pw_w2  = (const float*)d_in[23]; P.pw_b2  = (const float*)d_in[24];
  P.pw_w3  = (const float*)d_in[25]; P.pw_b3  = (const float*)d_in[26];
  P.gate_w1 = (const float*)d_in[27]; P.gate_b1 = (const float*)d_in[28];
  P.gate_w2 = (const float*)d_in[29]; P.gate_b2 = (const float*)d_in[30];
  P.out = (float*)d_out;

  dim3 grid(16384 / 16);  // 1024 tiles of 16 rows
  dim3 block(32);         // one wave32
  hipLaunchKernelGGL(awm_kernel, grid, block, 0, stream, P);
}